// CLIFSpike_36129264894720
// MI455X (gfx1250) — compile-verified
//
#include <hip/hip_runtime.h>

// CLIF spiking-neuron scan, [T=64, B=128, D=4096] f32 -> spikes f32.
// Elementwise recurrence over T: pure streaming-bandwidth kernel.
// 268 MB total traffic / 23.3 TB/s => ~11.5 us floor on MI455X.

typedef float v4f __attribute__((ext_vector_type(4)));

namespace {
constexpr int   T_STEPS = 64;
constexpr float GAMMA   = 0.5f;                     // 1 - 1/tau, tau = 2
constexpr float OMG     = 0.5f;                     // 1 - gamma
constexpr float VTH     = 1.0f;
constexpr float NLOG2E  = -1.44269504088896340736f; // -log2(e)
constexpr int   PIPE    = 4;                        // register pipeline depth
constexpr int   PF_DIST = 16;                       // L2 prefetch distance (timesteps)
}

// sigmoid(x) = 1/(1+exp(-x)) via hardware v_exp_f32 (2^x) + v_rcp_f32.
__device__ __forceinline__ float fast_sigmoid(float x) {
    return __builtin_amdgcn_rcpf(1.0f + __builtin_amdgcn_exp2f(x * NLOG2E));
}

// One CLIF timestep on 4 lanes held in registers. Reproduces the reference
// exactly, including the in-place sigmoid_ semantics (m_next = sigmoid(m)).
__device__ __forceinline__ v4f clif_step(const v4f xc, float (&u)[4], float (&m)[4]) {
    v4f s;
#pragma unroll
    for (int c = 0; c < 4; ++c) {
        float uu = fmaf(GAMMA, u[c], xc[c]);          // u = gamma*u + x_t
        float sp = (uu - VTH >= 0.0f) ? 1.0f : 0.0f;  // rect_spike(u - V_TH)
        float mm = fmaf(m[c], fast_sigmoid(OMG * uu), sp); // m = m*sig((1-g)u)+spike
        float sm = fast_sigmoid(mm);                  // in-place sigmoid_(m)
        u[c]     = fmaf(-sp, VTH + sm, uu);           // u -= spike*(V_TH + sm)
        m[c]     = sm;
        s[c]     = sp;
    }
    return s;
}

__global__ __launch_bounds__(256, 1) void clif_scan_kernel(
        const v4f* __restrict__ x, v4f* __restrict__ out, int n4) {
    const int i = blockIdx.x * blockDim.x + threadIdx.x;
    if (i >= n4) return;

    const v4f* xp = x + i;    // column through [T, B*D/4], stride n4 per step
    v4f*       op = out + i;

    float u[4] = {0.f, 0.f, 0.f, 0.f};
    float m[4] = {0.f, 0.f, 0.f, 0.f};

    // Prime a 4-deep register pipeline of non-temporal b128 loads.
    v4f buf[PIPE];
#pragma unroll
    for (int k = 0; k < PIPE; ++k)
        buf[k] = __builtin_nontemporal_load(xp + (size_t)k * n4);

    // Main loop: steps 0 .. T-PIPE-1, each iteration refills its pipeline slot
    // with the load for t+PIPE and prefetches t+PF_DIST into GL2.
#pragma unroll 1
    for (int tt = 0; tt + PIPE < T_STEPS; tt += PIPE) {
#pragma unroll
        for (int j = 0; j < PIPE; ++j) {
            const int t  = tt + j;
            const v4f xc = buf[j];
            buf[j] = __builtin_nontemporal_load(xp + (size_t)(t + PIPE) * n4);
            if (t + PF_DIST < T_STEPS)
                __builtin_prefetch((const void*)(xp + (size_t)(t + PF_DIST) * n4), 0, 0);
            const v4f s = clif_step(xc, u, m);
            __builtin_nontemporal_store(s, op + (size_t)t * n4);
        }
    }

    // Epilogue: last PIPE steps, pipeline already full, no more loads.
#pragma unroll
    for (int j = 0; j < PIPE; ++j) {
        const int t = T_STEPS - PIPE + j;
        const v4f s = clif_step(buf[j], u, m);
        __builtin_nontemporal_store(s, op + (size_t)t * n4);
    }
}

extern "C" void kernel_launch(void* const* d_in, const int* in_sizes, int n_in,
                              void* d_out, int out_size, void* d_ws, size_t ws_size,
                              hipStream_t stream) {
    (void)n_in; (void)out_size; (void)d_ws; (void)ws_size;
    const float* x   = (const float*)d_in[0];
    float*       out = (float*)d_out;

    const int total = in_sizes[0];        // T * B * D = 33,554,432
    const int n     = total / T_STEPS;    // B * D     = 524,288
    const int n4    = n / 4;              // float4 columns = 131,072

    const int block = 256;
    const int grid  = (n4 + block - 1) / block;
    clif_scan_kernel<<<grid, block, 0, stream>>>(
        (const v4f*)x, (v4f*)out, n4);
}